// GraphSage_4698694222361
// MI455X (gfx1250) — compile-verified
//
#include <hip/hip_runtime.h>
#include <hip/hip_bf16.h>

typedef __attribute__((ext_vector_type(2))) float v2f;
typedef __attribute__((ext_vector_type(8))) float v8f;

#define N_NODES 100000
#define N_EDGES 1600000
#define D_IN 64
#define D_HID 64
#define D_OUT 40

// ---------------------------------------------------------------- utilities
__global__ void zero_f32(float* __restrict__ p, long long n) {
  long long i = (long long)blockIdx.x * blockDim.x + threadIdx.x;
  long long stride = (long long)gridDim.x * blockDim.x;
  for (; i < n; i += stride) p[i] = 0.0f;
}

__global__ void degree_kernel(const int* __restrict__ dst,
                              float* __restrict__ deg) {
  int e = blockIdx.x * blockDim.x + threadIdx.x;
  if (e >= N_EDGES) return;
  unsigned d = (unsigned)dst[e];
  if (d < N_NODES) atomicAdd(&deg[d], 1.0f);
}

// thread per (edge, feature) element: coalesced gather + f32 atomic scatter.
// agg (25.6MB) fits in the 192MB L2, so atomics resolve at L2 bandwidth.
__global__ void scatter_kernel(const float* __restrict__ feat,
                               const int* __restrict__ src,
                               const int* __restrict__ dst,
                               float* __restrict__ agg) {
  long long i = (long long)blockIdx.x * blockDim.x + threadIdx.x;
  if (i >= (long long)N_EDGES * 64) return;
  int e = (int)(i >> 6);
  int c = (int)(i & 63);
  unsigned s = (unsigned)src[e];
  unsigned d = (unsigned)dst[e];
  if (s >= N_NODES || d >= N_NODES) return;  // safety clamp
  atomicAdd(&agg[(long long)d * 64 + c], feat[(long long)s * 64 + c]);
}

// ---------------------------------------------------------------- WMMA GEMM
// out[row, n] = sum_k  A[row,k] * B[k,n] + bias[n]
//   A = [ agg/max(deg,1) | self ]   (row-block of 16 nodes, K = 128)
//   B = [ Wl.T ; Wr.T ]             (staged transposed in LDS, stride 130)
// One wave computes a 16 x (NT*16) tile with V_WMMA_F32_16X16X4_F32.
template <int NT, bool RELU, int OUT_COLS>
__global__ void sage_linear_wmma(const float* __restrict__ agg,
                                 const float* __restrict__ deg,
                                 const float* __restrict__ self,
                                 const float* __restrict__ Wl,   // [OUT_COLS][64]
                                 const float* __restrict__ Wr,   // [OUT_COLS][64]
                                 const float* __restrict__ bias, // [OUT_COLS]
                                 float* __restrict__ out) {
  constexpr int NC = NT * 16;  // padded output columns
  __shared__ float ldsB[NC * 130];  // [n][k], stride 130 avoids bank conflicts
  __shared__ float ldsBias[NC];

  // cooperative stage of combined transposed weights: ldsB[n][k]
  for (int i = threadIdx.x; i < NC * 128; i += blockDim.x) {
    int n = i >> 7;
    int k = i & 127;
    float v = 0.0f;
    if (n < OUT_COLS) v = (k < 64) ? Wl[n * 64 + k] : Wr[n * 64 + (k - 64)];
    ldsB[n * 130 + k] = v;
  }
  for (int i = threadIdx.x; i < NC; i += blockDim.x)
    ldsBias[i] = (i < OUT_COLS) ? bias[i] : 0.0f;
  __syncthreads();

  const int lane = threadIdx.x & 31;
  const int wave = threadIdx.x >> 5;                // 4 waves / block
  const int rowbase = blockIdx.x * 64 + wave * 16;  // 16 rows per wave
  const int mrow = rowbase + (lane & 15);           // A-fragment row (M=lane%16)
  const int r = (mrow < N_NODES) ? mrow : (N_NODES - 1);  // clamp: EXEC stays full
  const int kk = (lane >> 4) << 1;                  // lanes 16..31 carry K+2,K+3

  const float rd = 1.0f / fmaxf(deg[r], 1.0f);
  const float* __restrict__ aggp = agg + (long long)r * 64;
  const float* __restrict__ selfp = self + (long long)r * 64;

  // accumulators pre-seeded with bias (bias depends only on column n)
  v8f acc[NT];
#pragma unroll
  for (int nt = 0; nt < NT; ++nt) {
    float bn = ldsBias[nt * 16 + (lane & 15)];
#pragma unroll
    for (int v = 0; v < 8; ++v) acc[nt][v] = bn;
  }

  // K = 128 in 32 steps of 4 (fp32 WMMA), 16x4 A-fragment = 2 VGPRs/lane
  for (int kt = 0; kt < 32; ++kt) {
    const int kb = kt * 4;
    v2f a;
    if (kt < 16) {  // wave-uniform branch: aggregated (mean) half of K
      a = *(const v2f*)(aggp + kb + kk);
      a.x *= rd;
      a.y *= rd;
    } else {        // self half of K
      a = *(const v2f*)(selfp + (kb - 64) + kk);
    }
#pragma unroll
    for (int nt = 0; nt < NT; ++nt) {
      const v2f b =
          *(const v2f*)(&ldsB[(nt * 16 + (lane & 15)) * 130 + kb + kk]);
      acc[nt] = __builtin_amdgcn_wmma_f32_16x16x4_f32(
          false, a, false, b, (short)0, acc[nt], false, false);
    }
  }

  // store 16x16 f32 C/D layout: VGPR v -> row v + 8*(lane>=16), col = lane%16
  const int colbase = lane & 15;
  const int rofs = (lane >> 4) << 3;
#pragma unroll
  for (int nt = 0; nt < NT; ++nt) {
#pragma unroll
    for (int v = 0; v < 8; ++v) {
      int row = rowbase + rofs + v;
      int col = nt * 16 + colbase;
      if (row < N_NODES && col < OUT_COLS) {
        float val = acc[nt][v];
        if (RELU) val = fmaxf(val, 0.0f);
        out[(long long)row * OUT_COLS + col] = val;
      }
    }
  }
}

// ---------------------------------------------------------------- softmax
__global__ void log_softmax_kernel(float* __restrict__ out) {
  int i = blockIdx.x * blockDim.x + threadIdx.x;
  if (i >= N_NODES) return;
  float v[D_OUT];
  float m = -3.402823466e38f;
#pragma unroll
  for (int c = 0; c < D_OUT; ++c) {
    v[c] = out[(long long)i * D_OUT + c];
    m = fmaxf(m, v[c]);
  }
  float s = 0.0f;
#pragma unroll
  for (int c = 0; c < D_OUT; ++c) s += __expf(v[c] - m);
  float ls = __logf(s);
#pragma unroll
  for (int c = 0; c < D_OUT; ++c)
    out[(long long)i * D_OUT + c] = v[c] - m - ls;
}

// ---------------------------------------------------------------- launcher
extern "C" void kernel_launch(void* const* d_in, const int* in_sizes, int n_in,
                              void* d_out, int out_size, void* d_ws,
                              size_t ws_size, hipStream_t stream) {
  const float* x   = (const float*)d_in[0];
  const int*   ei  = (const int*)d_in[1];   // edge_index [2, E]
  const float* W1l = (const float*)d_in[2];
  const float* b1  = (const float*)d_in[3];
  const float* W1r = (const float*)d_in[4];
  const float* W2l = (const float*)d_in[5];
  const float* b2  = (const float*)d_in[6];
  const float* W2r = (const float*)d_in[7];
  float* out = (float*)d_out;

  const int* src = ei;
  const int* dst = ei + N_EDGES;

  // workspace layout: deg[N] | agg[N*64] | h[N*64]   (~52 MB)
  float* deg = (float*)d_ws;
  float* agg = deg + N_NODES;
  float* h   = agg + (long long)N_NODES * 64;

  const int scatter_blocks = (int)(((long long)N_EDGES * 64 + 255) / 256);
  const int gemm_blocks = (N_NODES + 63) / 64;

  // ---- layer 1 ----
  zero_f32<<<2048, 256, 0, stream>>>(deg, (long long)N_NODES * (1 + 64));
  degree_kernel<<<(N_EDGES + 255) / 256, 256, 0, stream>>>(dst, deg);
  scatter_kernel<<<scatter_blocks, 256, 0, stream>>>(x, src, dst, agg);
  sage_linear_wmma<4, true, D_HID>
      <<<gemm_blocks, 128, 0, stream>>>(agg, deg, x, W1l, W1r, b1, h);

  // ---- layer 2 ----
  zero_f32<<<2048, 256, 0, stream>>>(agg, (long long)N_NODES * 64);
  scatter_kernel<<<scatter_blocks, 256, 0, stream>>>(h, src, dst, agg);
  sage_linear_wmma<3, false, D_OUT>
      <<<gemm_blocks, 128, 0, stream>>>(agg, deg, h, W2l, W2r, b2, out);

  // ---- log-softmax in place on d_out ----
  log_softmax_kernel<<<(N_NODES + 255) / 256, 256, 0, stream>>>(out);
}